// LightGCN_27238682591705
// MI455X (gfx1250) — compile-verified
//
#include <hip/hip_runtime.h>
#include <hip/hip_bf16.h>

// ---------------------------------------------------------------------------
// LightGCN on MI455X (gfx1250, wave32)
//   acc = e0; e_{l+1} = A e_l; acc += e_l ; out = acc/4
// SpMM is scatter-atomic bound; embeddings (76.8MB) are L2-resident (192MB L2).
// ---------------------------------------------------------------------------

typedef __attribute__((ext_vector_type(2))) float v2f;
typedef __attribute__((ext_vector_type(8))) float v8f;

#define NU 200000
#define NI 100000
#define DIM 64
#define NFLAT ((size_t)(NU + NI) * DIM)   // 19,200,000 floats
#define NV4   ((int)(NFLAT / 4))          // 4,800,000 float4
#define NTILES ((int)(NFLAT / 256))       // 75,000 WMMA tiles

// init: acc (d_out) = all_emb ; emb_cur (ws0) = all_emb
__global__ void __launch_bounds__(256)
lgcn_init(const float4* __restrict__ user4, const float4* __restrict__ item4,
          float4* __restrict__ acc4, float4* __restrict__ cur4,
          int n_user4, int n4) {
  int i = blockIdx.x * 256 + threadIdx.x;
  if (i >= n4) return;
  float4 v = (i < n_user4) ? user4[i] : item4[i - n_user4];
  acc4[i] = v;
  cur4[i] = v;
}

// SpMM scatter: y[rows[e]] += vals[e] * x[cols[e]]   (y pre-initialized)
// 16 lanes per edge, float4 per lane (row = 256B contiguous gather).
__global__ void __launch_bounds__(256)
lgcn_scatter(const float* __restrict__ vals, const int* __restrict__ rows,
             const int* __restrict__ cols, const float* __restrict__ x,
             float* __restrict__ y, int n_edges) {
  unsigned gid = blockIdx.x * 256u + threadIdx.x;
  int e = (int)(gid >> 4);
  if (e >= n_edges) return;
  int l = (int)(gid & 15u);

  if (l == 0) {
    // keep the streaming edge arrays flowing; gfx1250 global_prefetch_b8
    __builtin_prefetch(cols + e + 4096, 0, 0);
    __builtin_prefetch(rows + e + 4096, 0, 0);
    __builtin_prefetch(vals + e + 4096, 0, 0);
  }

  // read-once streams: non-temporal so they don't evict L2-resident x/y
  int   col = __builtin_nontemporal_load(cols + e);
  int   row = __builtin_nontemporal_load(rows + e);
  float v   = __builtin_nontemporal_load(vals + e);

  const float4* xr = (const float4*)(x + (size_t)col * DIM);
  float4 g = xr[l];                       // global_load_b128, L2-resident

  float* yr = y + (size_t)row * DIM + l * 4;
  atomicAdd(yr + 0, v * g.x);             // global_atomic_add_f32 (no return)
  atomicAdd(yr + 1, v * g.y);
  atomicAdd(yr + 2, v * g.z);
  atomicAdd(yr + 3, v * g.w);
}

// acc += src
__global__ void __launch_bounds__(256)
lgcn_add(float4* __restrict__ acc4, const float4* __restrict__ src4, int n4) {
  int i = blockIdx.x * 256 + threadIdx.x;
  if (i >= n4) return;
  float4 a = acc4[i], s = src4[i];
  a.x += s.x; a.y += s.y; a.z += s.z; a.w += s.w;
  acc4[i] = a;
}

// Final epilogue: data *= 0.25, done per 256-float tile as a 16x16 matrix T:
//   D = sum_j A_j x B_j, A_j = T[:,4j:4j+4], B_j = 0.25 * shifted identity
// => D = 0.25 * T exactly.  Uses v_wmma_f32_16x16x4_f32, staged via LDS so
// global loads/stores stay fully coalesced (float4 per lane).
__global__ void __launch_bounds__(256)
lgcn_scale_wmma(float* __restrict__ data, int ntiles) {
  __shared__ float lds[8 * 256];
  const int lane = threadIdx.x & 31;
  const int wave = threadIdx.x >> 5;
  const int tile = blockIdx.x * 8 + wave;
  const bool active = (tile < ntiles);

  float*  t  = lds + wave * 256;
  float4* t4 = (float4*)t;
  float4* g4 = (float4*)data + (size_t)(active ? tile : 0) * 64;

  if (active) {                 // coalesced tile load -> LDS
    t4[lane * 2 + 0] = g4[lane * 2 + 0];
    t4[lane * 2 + 1] = g4[lane * 2 + 1];
  }
  __syncthreads();

  if (active) {
    const int m    = lane & 15;   // row (A) / column (B,D) index
    const int half = lane >> 4;   // K-pair select per ISA 16x4 f32 layout
    v8f c = {0.f, 0.f, 0.f, 0.f, 0.f, 0.f, 0.f, 0.f};
#pragma unroll
    for (int j = 0; j < 4; ++j) {
      // A_j layout: vgpr0 = K(2*half), vgpr1 = K(2*half+1), M = lane%16
      v2f a;
      a.x = t[m * 16 + 4 * j + 2 * half + 0];
      a.y = t[m * 16 + 4 * j + 2 * half + 1];
      // B_j[k][n] = 0.25 * (n == 4j+k); vgpr0 = row 2*half, vgpr1 = row 2*half+1
      v2f b;
      b.x = (m == 4 * j + 2 * half + 0) ? 0.25f : 0.0f;
      b.y = (m == 4 * j + 2 * half + 1) ? 0.25f : 0.0f;
      c = __builtin_amdgcn_wmma_f32_16x16x4_f32(
          /*neg_a=*/false, a, /*neg_b=*/false, b,
          /*c_mod=*/(short)0, c, /*reuse_a=*/false, /*reuse_b=*/false);
    }
    // D layout: vgpr r -> M = r + 8*half, N = m
#pragma unroll
    for (int r = 0; r < 8; ++r)
      t[(r + 8 * half) * 16 + m] = c[r];
  }
  __syncthreads();

  if (active) {                 // coalesced tile store
    g4[lane * 2 + 0] = t4[lane * 2 + 0];
    g4[lane * 2 + 1] = t4[lane * 2 + 1];
  }
}

extern "C" void kernel_launch(void* const* d_in, const int* in_sizes, int n_in,
                              void* d_out, int out_size, void* d_ws, size_t ws_size,
                              hipStream_t stream) {
  const float* user_emb = (const float*)d_in[0];
  const float* item_emb = (const float*)d_in[1];
  const float* adj_vals = (const float*)d_in[2];
  const int*   adj_rows = (const int*)d_in[3];
  const int*   adj_cols = (const int*)d_in[4];
  float* acc = (float*)d_out;

  const int E = in_sizes[2];
  float* ws0 = (float*)d_ws;         // emb buffers, 76.8MB each
  float* ws1 = ws0 + NFLAT;

  const int gb_v4 = (NV4 + 255) / 256;
  const int gb_sc = (int)(((long long)E * 16 + 255) / 256);

  // acc = all_emb ; cur = all_emb
  lgcn_init<<<gb_v4, 256, 0, stream>>>((const float4*)user_emb,
                                       (const float4*)item_emb,
                                       (float4*)acc, (float4*)ws0,
                                       NU * DIM / 4, NV4);
  // layer 1: ws1 = A*ws0 ; acc += ws1
  hipMemsetAsync(ws1, 0, NFLAT * sizeof(float), stream);
  lgcn_scatter<<<gb_sc, 256, 0, stream>>>(adj_vals, adj_rows, adj_cols, ws0, ws1, E);
  lgcn_add<<<gb_v4, 256, 0, stream>>>((float4*)acc, (const float4*)ws1, NV4);
  // layer 2: ws0 = A*ws1 ; acc += ws0
  hipMemsetAsync(ws0, 0, NFLAT * sizeof(float), stream);
  lgcn_scatter<<<gb_sc, 256, 0, stream>>>(adj_vals, adj_rows, adj_cols, ws1, ws0, E);
  lgcn_add<<<gb_v4, 256, 0, stream>>>((float4*)acc, (const float4*)ws0, NV4);
  // layer 3: scatter-add directly into acc (saves one zero + one add pass)
  lgcn_scatter<<<gb_sc, 256, 0, stream>>>(adj_vals, adj_rows, adj_cols, ws0, acc, E);
  // acc *= 0.25 via WMMA tiles
  lgcn_scale_wmma<<<(NTILES + 7) / 8, 256, 0, stream>>>(acc, NTILES);
}